// GNN_10170482557309
// MI455X (gfx1250) — compile-verified
//
#include <hip/hip_runtime.h>

typedef __attribute__((ext_vector_type(2))) float v2f;
typedef __attribute__((ext_vector_type(8))) float v8f;

#define HID 64

// ---------------- degree / norm ----------------

__global__ void k_init_deg(float* __restrict__ deg, int n) {
  int i = blockIdx.x * blockDim.x + threadIdx.x;
  if (i < n) deg[i] = 1.0f;  // self loop
}

__global__ void k_edge_deg(const long long* __restrict__ edges, float* __restrict__ deg, int E) {
  int e = blockIdx.x * blockDim.x + threadIdx.x;
  if (e < E) {
    int dst = (int)edges[2 * (long long)e + 1];
    atomicAdd(&deg[dst], 1.0f);
  }
}

__global__ void k_rsqrt(float* __restrict__ dinv, int n) {
  int i = blockIdx.x * blockDim.x + threadIdx.x;
  if (i < n) dinv[i] = rsqrtf(dinv[i]);
}

// ---------------- WMMA GEMM: C[M,N] = A[M,K] @ B[K,N] (+bias) ----------------
// Block = 4 waves (128 threads). Block tile: 128 rows x 64 cols.
// Each wave: 32 rows x 64 cols = 8 accumulators (v8f).
// B is staged per 64-deep K-chunk into LDS in WMMA fragment layout so the
// inner loop reads B with one ds_load_b64 per 16x16 tile.
//
// f32 WMMA 16x16x4 fragment layouts (ISA 7.12.2):
//   A (16x4): lane L holds row M=L&15; VGPR0 = K=2*(L>=16), VGPR1 = K+1
//   B (4x16): lane L holds col N=L&15; VGPR0 = K=2*(L>=16), VGPR1 = K+1
//   D (16x16): VGPR r = row r (lanes 0-15) / row r+8 (lanes 16-31), col = L&15
__global__ __launch_bounds__(128) void k_gemm_wmma(
    const float* __restrict__ A, const float* __restrict__ B,
    const float* __restrict__ bias, float* __restrict__ C,
    int M, int N, int K) {
  __shared__ float Bs[4096];  // 16 KB: one K-chunk (64) x 64 cols, fragment layout

  const int tid = threadIdx.x;
  const int lane = tid & 31;
  const int wave = tid >> 5;
  const int nStrips = N >> 6;
  const int mb = blockIdx.x / nStrips;
  const int strip = blockIdx.x % nStrips;
  const int n0 = strip * 64;
  const int r0 = mb * 128 + wave * 32;

  const int hi = lane >> 4;  // 0/1: K half within fragment
  const int lo = lane & 15;  // row (A) / col (B,D)

  // Clamp A row pointers so tail waves stay barrier-alive with in-range loads.
  int rowa = r0 + lo;      if (rowa >= M) rowa = M - 1;
  int rowb = r0 + 16 + lo; if (rowb >= M) rowb = M - 1;
  const float* __restrict__ arowa = A + (size_t)rowa * K + 2 * hi;
  const float* __restrict__ arowb = A + (size_t)rowb * K + 2 * hi;

  v8f acc[8];
#pragma unroll
  for (int t = 0; t < 8; ++t) acc[t] = (v8f){};

  for (int k0 = 0; k0 < K; k0 += 64) {
    __syncthreads();  // previous chunk's reads done before refill
    // Cooperative fill: element B[k0+kk][n0+col] -> fragment slot
    //   step=kk>>2, tile=col>>4, lane=(col&15)|(((kk>>1)&1)<<4), slot=kk&1
    //   float index = ((step*4+tile)*32 + lane)*2 + slot
#pragma unroll
    for (int j = 0; j < 32; ++j) {
      int flat = tid + 128 * j;  // 4096 elements total
      int kk = flat >> 6;
      int col = flat & 63;
      float v = B[(size_t)(k0 + kk) * N + n0 + col];
      int l = (col & 15) | (((kk >> 1) & 1) << 4);
      int idx = (((kk >> 2) * 4 + (col >> 4)) * 32 + l) * 2 + (kk & 1);
      Bs[idx] = v;
    }
    __syncthreads();

#pragma unroll
    for (int s = 0; s < 16; ++s) {
      v2f a0 = *(const v2f*)(arowa + k0 + 4 * s);
      v2f a1 = *(const v2f*)(arowb + k0 + 4 * s);
      const v2f* bs = ((const v2f*)Bs) + s * 128 + lane;
#pragma unroll
      for (int t = 0; t < 4; ++t) {
        v2f b = bs[t * 32];  // ds_load_b64
        acc[t]     = __builtin_amdgcn_wmma_f32_16x16x4_f32(false, a0, false, b, (short)0, acc[t],     false, false);
        acc[4 + t] = __builtin_amdgcn_wmma_f32_16x16x4_f32(false, a1, false, b, (short)0, acc[4 + t], false, false);
      }
    }
  }

  float bia0 = 0.f, bia1 = 0.f, bia2 = 0.f, bia3 = 0.f;
  if (bias) {
    bia0 = bias[n0 + lo];
    bia1 = bias[n0 + 16 + lo];
    bia2 = bias[n0 + 32 + lo];
    bia3 = bias[n0 + 48 + lo];
  }
#pragma unroll
  for (int g = 0; g < 2; ++g) {
#pragma unroll
    for (int r = 0; r < 8; ++r) {
      int row = r0 + 16 * g + 8 * hi + r;
      if (row < M) {
        float* __restrict__ crow = C + (size_t)row * N + n0 + lo;
        crow[0]  = acc[g * 4 + 0][r] + bia0;
        crow[16] = acc[g * 4 + 1][r] + bia1;
        crow[32] = acc[g * 4 + 2][r] + bia2;
        crow[48] = acc[g * 4 + 3][r] + bia3;
      }
    }
  }
}

// ---------------- aggregation helpers ----------------

// g[i,c] = h[i,c]*dinv[i]; acc[i,c] = g[i,c]  (self-loop init)
__global__ void k_scale_copy(const float* __restrict__ h, const float* __restrict__ dinv,
                             float* __restrict__ g, float* __restrict__ acc, int n) {
  long long t = (long long)blockIdx.x * blockDim.x + threadIdx.x;
  if (t < (long long)n * HID) {
    int i = (int)(t >> 6);
    float v = h[t] * dinv[i];
    g[t] = v;
    acc[t] = v;
  }
}

// acc[dst] += g[src] for every edge (64 lanes per edge)
__global__ void k_edge_agg(const long long* __restrict__ edges, const float* __restrict__ g,
                           float* __restrict__ acc, int E) {
  long long t = (long long)blockIdx.x * blockDim.x + threadIdx.x;
  if (t < (long long)E * HID) {
    int e = (int)(t >> 6);
    int c = (int)(t & 63);
    int src = (int)edges[2 * (long long)e];
    int dst = (int)edges[2 * (long long)e + 1];
    atomicAdd(&acc[(size_t)dst * HID + c], g[(size_t)src * HID + c]);
  }
}

// out[i,c] = maybe_relu(dinv[i]*acc[i,c] + bias[c])
__global__ void k_finish(const float* __restrict__ acc, const float* __restrict__ dinv,
                         const float* __restrict__ bias, float* __restrict__ out,
                         int n, int relu) {
  long long t = (long long)blockIdx.x * blockDim.x + threadIdx.x;
  if (t < (long long)n * HID) {
    int i = (int)(t >> 6);
    int c = (int)(t & 63);
    float v = dinv[i] * acc[t];
    if (bias) v += bias[c];
    if (relu) v = fmaxf(v, 0.0f);
    out[t] = v;
  }
}

// ---------------- driver ----------------

extern "C" void kernel_launch(void* const* d_in, const int* in_sizes, int n_in,
                              void* d_out, int out_size, void* d_ws, size_t ws_size,
                              hipStream_t stream) {
  const float* x  = (const float*)d_in[0];             // 50000 x 640
  const long long* edges = (const long long*)d_in[1];  // 200000 x 2, int64
  const float* W1 = (const float*)d_in[2];             // 640 x 64
  const float* b1 = (const float*)d_in[3];
  const float* W2 = (const float*)d_in[4];             // 64 x 64
  const float* b2 = (const float*)d_in[5];
  const float* W3 = (const float*)d_in[6];             // 64 x 640
  const float* b3 = (const float*)d_in[7];
  float* out = (float*)d_out;

  const int n = in_sizes[0] / 640;  // 50000
  const int E = in_sizes[1] / 2;    // 200000
  const int Din = 640;

  float* ws   = (float*)d_ws;
  float* dinv = ws;                      // n (deg -> dinv in place)
  float* bufA = ws + 50048;              // n*64
  float* bufB = bufA + (size_t)n * HID;  // n*64
  float* bufC = bufB + (size_t)n * HID;  // n*64

  const int T = 256;
  const int nodeBlk = (n + T - 1) / T;
  const int edgeBlk = (E + T - 1) / T;
  const int featBlk = (int)(((long long)n * HID + T - 1) / T);
  const int eAggBlk = (int)(((long long)E * HID + T - 1) / T);
  auto gemmBlk = [&](int N_) { return ((n + 127) / 128) * (N_ / 64); };

  // degree -> dinv
  k_init_deg<<<nodeBlk, T, 0, stream>>>(dinv, n);
  k_edge_deg<<<edgeBlk, T, 0, stream>>>(edges, dinv, E);
  k_rsqrt<<<nodeBlk, T, 0, stream>>>(dinv, n);

  // ---- layer 1: H1 = relu(Agg(X@W1) + b1) ----
  k_gemm_wmma<<<gemmBlk(HID), 128, 0, stream>>>(x, W1, nullptr, bufA, n, HID, Din);
  k_scale_copy<<<featBlk, T, 0, stream>>>(bufA, dinv, bufB, bufC, n);  // g=bufB acc=bufC
  k_edge_agg<<<eAggBlk, T, 0, stream>>>(edges, bufB, bufC, E);
  k_finish<<<featBlk, T, 0, stream>>>(bufC, dinv, b1, bufA, n, 1);     // H1=bufA

  // ---- layer 2: H2 = relu(Agg(H1@W2) + b2) ----
  k_gemm_wmma<<<gemmBlk(HID), 128, 0, stream>>>(bufA, W2, nullptr, bufB, n, HID, HID);
  k_scale_copy<<<featBlk, T, 0, stream>>>(bufB, dinv, bufC, bufA, n);  // g=bufC acc=bufA
  k_edge_agg<<<eAggBlk, T, 0, stream>>>(edges, bufC, bufA, E);
  k_finish<<<featBlk, T, 0, stream>>>(bufA, dinv, b2, bufB, n, 1);     // H2=bufB

  // ---- layer 3 (reordered Agg/GEMM): Out = Agg(H2) @ W3 + b3 ----
  k_scale_copy<<<featBlk, T, 0, stream>>>(bufB, dinv, bufC, bufA, n);  // g=bufC acc=bufA
  k_edge_agg<<<eAggBlk, T, 0, stream>>>(edges, bufC, bufA, E);
  k_finish<<<featBlk, T, 0, stream>>>(bufA, dinv, nullptr, bufC, n, 0);  // Agg(H2)=bufC
  k_gemm_wmma<<<gemmBlk(Din), 128, 0, stream>>>(bufC, W3, b3, out, n, Din, HID);
}